// LIFLayer_10617159156206
// MI455X (gfx1250) — compile-verified
//
#include <hip/hip_runtime.h>

// LIF layer for MI455X (gfx1250, wave32).
// Phase 1: Wx = x[64000,256] * W[256,256]^T via V_WMMA_F32_16X16X4_F32 (exact fp32).
//          W tile (64x256) staged into LDS once per block with the Tensor Data
//          Mover (tensor_load_to_lds), using TDM padding for conflict-free banks.
//          Wx written directly into d_out (same 64*1000*256 f32 footprint as the
//          output; stays resident in the 192MB L2 between phases).
// Phase 2: in-place leaky-integrate-and-fire scan over T, one thread per (b,h).

typedef __attribute__((ext_vector_type(2))) float        v2f;
typedef __attribute__((ext_vector_type(8))) float        v8f;
typedef __attribute__((ext_vector_type(4))) unsigned int v4u;
typedef __attribute__((ext_vector_type(8))) int          v8i;
typedef __attribute__((ext_vector_type(4))) int          v4i;

#define ALPHA_MIN 0.8187307530779818f
#define ALPHA_MAX 0.9607894391523232f

constexpr int Bsz = 64;
constexpr int T   = 1000;
constexpr int I   = 256;
constexpr int H   = 256;
constexpr int M   = Bsz * T;        // 64000 rows
constexpr int LDS_STRIDE = 260;     // 256 dwords/row + 4 dwords TDM pad (bank-conflict-free)

// ---------------------------------------------------------------------------
// Phase 1: WMMA GEMM.  Block = 4 waves; wave w computes M-tile (blockIdx.x*4+w),
// covering 16 rows x 64 columns (4 accumulators) with A-fragment reuse.
// All 4 waves share one 64-row W tile, DMA'd to LDS by the TDM.
//
// fp32 WMMA operand layout (ISA 7.12.2, 32-bit A 16x4):
//   lane L (L<16):  row M=L, VGPR0..1 = K={0,1};  lane L+16: K={2,3}
// B (4x16) symmetric: lane holds column N=L%16 (= row of W), K pair by L/16.
// ---------------------------------------------------------------------------
__global__ __launch_bounds__(128)
void lif_gemm_wmma(const float* __restrict__ x,
                   const float* __restrict__ W,
                   float* __restrict__ out)
{
    __shared__ float ldsW[64 * LDS_STRIDE];     // 66,560 B

    const int wave  = threadIdx.x >> 5;
    const int lane  = threadIdx.x & 31;
    const int half  = lane >> 4;                // selects K pair {0,1} vs {2,3}
    const int lr    = lane & 15;                // row (A) / column (B,C) within tile
    const int mbase = (blockIdx.x * 4 + wave) * 16;
    const int hbase = blockIdx.y * 64;

    // ---- TDM: DMA W[hbase:hbase+64, 0:256] (fp32) into LDS, padded rows. ----
    if (threadIdx.x < 32) {
        const unsigned long long ga =
            (unsigned long long)(const void*)(W + (size_t)hbase * I);
        const unsigned int lds_base = (unsigned int)(size_t)(void*)&ldsW[0];

        v4u g0;
        g0[0] = 1u;                                    // count=1 (valid), user mode
        g0[1] = lds_base;                              // lds_addr (bytes)
        g0[2] = (unsigned int)ga;                      // global_addr[31:0]
        g0[3] = (unsigned int)((ga >> 32) & 0x01FFFFFFu)
              | (2u << 30);                            // global_addr[56:32] | type=2

        v8i g1;
        g1[0] = (int)((2u << 16)                       // data_size = 4 bytes
                    | (1u << 20)                       // pad_enable
                    | (7u << 22)                       // pad_interval: 256 dwords (1 row)
                    | (3u << 25));                     // pad_amount: 4 dwords -> stride 260
        g1[1] = (int)(256u << 16);                     // tensor_dim0 = 256
        g1[2] = (int)( 64u << 16);                     // tensor_dim1 = 64 (tile-exact, no OOB)
        g1[3] = (int)(256u << 16);                     // tile_dim0 = 256
        g1[4] = 64;                                    // tile_dim1 = 64
        g1[5] = 256;                                   // tensor_dim0_stride = 256 elems
        g1[6] = 0;
        g1[7] = 0;

        v4i gz4 = {0, 0, 0, 0};                        // groups 2/3 unused (2D tensor)
        v8i gz8 = {0, 0, 0, 0, 0, 0, 0, 0};

        // amdgpu-toolchain (clang-23) 6-arg form.
        __builtin_amdgcn_tensor_load_to_lds(g0, g1, gz4, gz4, gz8, 0);
        __builtin_amdgcn_s_wait_tensorcnt(0);          // only issuing wave may wait
    }
    __syncthreads();

    // Per-lane fragment pointers: A from global (streaming), B from LDS.
    const float* arow = x + (size_t)(mbase + lr) * I + 2 * half;
    const float* lb0  = ldsW + (size_t)(lr +  0) * LDS_STRIDE + 2 * half;
    const float* lb1  = ldsW + (size_t)(lr + 16) * LDS_STRIDE + 2 * half;
    const float* lb2  = ldsW + (size_t)(lr + 32) * LDS_STRIDE + 2 * half;
    const float* lb3  = ldsW + (size_t)(lr + 48) * LDS_STRIDE + 2 * half;

    v8f c0 = {}, c1 = {}, c2 = {}, c3 = {};

#pragma unroll 4
    for (int k = 0; k < I; k += 4) {
        v2f a  = *(const v2f*)(arow + k);
        v2f b0 = *(const v2f*)(lb0  + k);
        v2f b1 = *(const v2f*)(lb1  + k);
        v2f b2 = *(const v2f*)(lb2  + k);
        v2f b3 = *(const v2f*)(lb3  + k);
        c0 = __builtin_amdgcn_wmma_f32_16x16x4_f32(false, a, false, b0,
                                                   (short)0, c0, false, false);
        c1 = __builtin_amdgcn_wmma_f32_16x16x4_f32(false, a, false, b1,
                                                   (short)0, c1, false, false);
        c2 = __builtin_amdgcn_wmma_f32_16x16x4_f32(false, a, false, b2,
                                                   (short)0, c2, false, false);
        c3 = __builtin_amdgcn_wmma_f32_16x16x4_f32(false, a, false, b3,
                                                   (short)0, c3, false, false);
    }

    // C/D layout: VGPR r, lane L -> (row = r + 8*(L/16), col = L%16).
    float* o = out + (size_t)(mbase + 8 * half) * H + hbase + lr;
#pragma unroll
    for (int r = 0; r < 8; ++r) {
        o[(size_t)r * H +  0] = c0[r];
        o[(size_t)r * H + 16] = c1[r];
        o[(size_t)r * H + 32] = c2[r];
        o[(size_t)r * H + 48] = c3[r];
    }
}

// ---------------------------------------------------------------------------
// Phase 2: in-place LIF scan.  Thread (b,h) walks out[b, :, h] (stride H,
// h-coalesced), replacing Wx with spikes.  Unroll x4 with hoisted loads so the
// memory stream runs ahead of the dependent FMA chain; prefetch 2 iterations out.
// ---------------------------------------------------------------------------
__global__ __launch_bounds__(256)
void lif_scan(const float* __restrict__ alpha,
              const float* __restrict__ u0,
              const float* __restrict__ s0,
              float* __restrict__ out)
{
    const int b = blockIdx.x;
    const int h = threadIdx.x;

    const float a  = fminf(fmaxf(alpha[h], ALPHA_MIN), ALPHA_MAX);
    const float oa = 1.0f - a;
    float u = u0[b * H + h];
    float s = s0[b * H + h];

    float* p = out + (size_t)b * T * H + h;

    for (int t = 0; t < T; t += 4) {
        const float w0 = p[0 * H];
        const float w1 = p[1 * H];
        const float w2 = p[2 * H];
        const float w3 = p[3 * H];
        __builtin_prefetch(p + 8 * H, 0, 1);   // global_prefetch_b8

        u = fmaf(a, u - s, oa * w0); s = (u > 1.0f) ? 1.0f : 0.0f; p[0 * H] = s;
        u = fmaf(a, u - s, oa * w1); s = (u > 1.0f) ? 1.0f : 0.0f; p[1 * H] = s;
        u = fmaf(a, u - s, oa * w2); s = (u > 1.0f) ? 1.0f : 0.0f; p[2 * H] = s;
        u = fmaf(a, u - s, oa * w3); s = (u > 1.0f) ? 1.0f : 0.0f; p[3 * H] = s;

        p += 4 * H;
    }
}

// ---------------------------------------------------------------------------
extern "C" void kernel_launch(void* const* d_in, const int* in_sizes, int n_in,
                              void* d_out, int out_size, void* d_ws, size_t ws_size,
                              hipStream_t stream)
{
    (void)in_sizes; (void)n_in; (void)out_size; (void)d_ws; (void)ws_size;

    const float* x     = (const float*)d_in[0];  // [B,T,I]
    const float* W     = (const float*)d_in[1];  // [H,I]
    const float* alpha = (const float*)d_in[2];  // [H]
    const float* u0    = (const float*)d_in[3];  // [B,H]
    const float* s0    = (const float*)d_in[4];  // [B,H]
    float* out = (float*)d_out;                  // [B,T,H]

    // Phase 1: 1000 x 4 blocks of 128 threads (4 waves), wave = one 16x64 tile.
    dim3 g1(M / (16 * 4), H / 64);
    lif_gemm_wmma<<<g1, 128, 0, stream>>>(x, W, out);

    // Phase 2: one thread per (b,h); 64 blocks x 256 threads.
    lif_scan<<<dim3(Bsz), 256, 0, stream>>>(alpha, u0, s0, out);
}